// GroupedQueryAttention_61117384622338
// MI455X (gfx1250) — compile-verified
//
#include <hip/hip_runtime.h>

// ---------------------------------------------------------------------------
// GroupedQueryAttention for MI455X (gfx1250), fp32 throughout via
// V_WMMA_F32_16X16X4_F32.  B=2, S=2048, DM=1024, H=16, KVH=4, DK=64.
//
// Pipeline:
//   1. Q = query @ Wq^T + bq          (WMMA NT gemm)
//   2. Kp = key @ Wk^T + bk, Vp = value @ Wv^T + bv
//   3. scores = Q_h @ K_g^T * 1/8  -> written raw into attn region of d_out
//   4. row softmax in-place on attn
//   5. ctx = attn @ V_g  (WMMA NN gemm), stored in [b,s,H*DK] layout
//   6. out = ctx @ Wo^T + bo
//
// GEMM core: block = 256 threads (8 waves), block tile 128x64, each wave
// computes a 32x32 sub-tile as 2x2 WMMA tiles (4 accumulators) so each
// ds_load_b64 fragment feeds two WMMAs (LDS-bandwidth balanced).
// ---------------------------------------------------------------------------

typedef __attribute__((ext_vector_type(2))) float v2f;
typedef __attribute__((ext_vector_type(4))) float v4f;
typedef __attribute__((ext_vector_type(8))) float v8f;

namespace cfg {
constexpr int B   = 2;
constexpr int S   = 2048;
constexpr int DM  = 1024;
constexpr int H   = 16;
constexpr int KVH = 4;
constexpr int DK  = 64;          // DM / H
constexpr int KVD = 256;         // DK * KVH
constexpr int REP = 4;           // H / KVH

constexpr int BM   = 128;        // block tile rows
constexpr int BN   = 64;         // block tile cols
constexpr int KC   = 32;         // K chunk staged in LDS
constexpr int KPAD = 36;         // padded LDS row stride (bank-conflict free)
}

// ---------------------------------------------------------------------------
// WMMA GEMM core.  BT=true : C = alpha * A[M,K] @ B[N,K]^T (+bias)
//                  BT=false: C = alpha * A[M,K] @ B[K,N]   (+bias)
// Wave w: rows (w&3)*32..+31, cols (w>>2)*32..+31 of the block tile.
// ---------------------------------------------------------------------------
template <bool BT>
__device__ __forceinline__ void gemm_core(const float* __restrict__ A, int lda,
                                          const float* __restrict__ Bm, int ldb,
                                          const float* __restrict__ bias,
                                          float* __restrict__ C, int ldc,
                                          int K, float alpha,
                                          int tileM, int tileN)
{
    using namespace cfg;
    __shared__ float As[BM * KPAD];   // 128 x 36 floats = 18 KB
    __shared__ float Bs[BN * KPAD];   //  64 x 36 floats =  9 KB

    const int tid  = threadIdx.x;
    const int wave = tid >> 5;
    const int lane = tid & 31;
    const int wm   = wave & 3;       // 0..3 : wave row group (x32)
    const int wn   = wave >> 2;      // 0..1 : wave col group (x32)
    const int l    = lane & 15;
    const int half = lane >> 4;

    v8f acc00 = {}, acc01 = {}, acc10 = {}, acc11 = {};

    // global staging coordinates (per thread)
    const int arow = tid >> 1;             // 0..127
    const int acol = (tid & 1) << 4;       // 0 or 16  (16 floats / thread)
    const int brow = tid >> 2;             // 0..63    (NT)
    const int bcol = (tid & 3) << 3;       // 0..24    (NT, 8 floats / thread)
    const int bc   = tid >> 3;             // 0..31    (NN depth)
    const int bn   = (tid & 7) << 3;       // 0..56    (NN, 8 floats / thread)

    for (int k0 = 0; k0 < K; k0 += KC) {
        __syncthreads();
        // ---- stage A tile: BM x KC (4096 floats, 16 per thread) ----
        {
            const float* src = A + (size_t)(tileM + arow) * lda + k0 + acol;
            v4f x0 = *(const v4f*)(src);
            v4f x1 = *(const v4f*)(src + 4);
            v4f x2 = *(const v4f*)(src + 8);
            v4f x3 = *(const v4f*)(src + 12);
            float* dst = &As[arow * KPAD + acol];
            *(v4f*)(dst)      = x0;
            *(v4f*)(dst + 4)  = x1;
            *(v4f*)(dst + 8)  = x2;
            *(v4f*)(dst + 12) = x3;
            if (k0 + KC < K) __builtin_prefetch(src + KC, 0, 0);
        }
        // ---- stage B tile: BN x KC (2048 floats, 8 per thread) ----
        if (BT) {
            // B is [N,K] row-major: Bs[n][c] = B[tileN+n][k0+c]
            const float* src = Bm + (size_t)(tileN + brow) * ldb + k0 + bcol;
            v4f x0 = *(const v4f*)(src);
            v4f x1 = *(const v4f*)(src + 4);
            float* dst = &Bs[brow * KPAD + bcol];
            *(v4f*)(dst)     = x0;
            *(v4f*)(dst + 4) = x1;
            if (k0 + KC < K) __builtin_prefetch(src + KC, 0, 0);
        } else {
            // B is [K,N] row-major: Bs[n][c] = B[k0+c][tileN+n]
            const float* src = Bm + (size_t)(k0 + bc) * ldb + tileN + bn;
            v4f x0 = *(const v4f*)(src);
            v4f x1 = *(const v4f*)(src + 4);
            Bs[(bn + 0) * KPAD + bc] = x0[0];
            Bs[(bn + 1) * KPAD + bc] = x0[1];
            Bs[(bn + 2) * KPAD + bc] = x0[2];
            Bs[(bn + 3) * KPAD + bc] = x0[3];
            Bs[(bn + 4) * KPAD + bc] = x1[0];
            Bs[(bn + 5) * KPAD + bc] = x1[1];
            Bs[(bn + 6) * KPAD + bc] = x1[2];
            Bs[(bn + 7) * KPAD + bc] = x1[3];
            if (k0 + KC < K) __builtin_prefetch(Bm + (size_t)(k0 + KC + bc) * ldb + tileN + bn, 0, 0);
        }
        __syncthreads();

        // ---- 8 K-steps x 4 WMMAs (2x2 register blocking) ----
        const float* __restrict__ a0p = &As[(wm * 32 + l) * KPAD];
        const float* __restrict__ a1p = &As[(wm * 32 + 16 + l) * KPAD];
        const float* __restrict__ b0p = &Bs[(wn * 32 + l) * KPAD];
        const float* __restrict__ b1p = &Bs[(wn * 32 + 16 + l) * KPAD];
#pragma unroll
        for (int kk = 0; kk < KC; kk += 4) {
            const int c = kk + (half << 1);
            v2f a0, a1, b0, b1;
            a0[0] = a0p[c]; a0[1] = a0p[c + 1];
            a1[0] = a1p[c]; a1[1] = a1p[c + 1];
            b0[0] = b0p[c]; b0[1] = b0p[c + 1];
            b1[0] = b1p[c]; b1[1] = b1p[c + 1];
            acc00 = __builtin_amdgcn_wmma_f32_16x16x4_f32(false, a0, false, b0,
                                                          (short)0, acc00, false, false);
            acc01 = __builtin_amdgcn_wmma_f32_16x16x4_f32(false, a0, false, b1,
                                                          (short)0, acc01, false, false);
            acc10 = __builtin_amdgcn_wmma_f32_16x16x4_f32(false, a1, false, b0,
                                                          (short)0, acc10, false, false);
            acc11 = __builtin_amdgcn_wmma_f32_16x16x4_f32(false, a1, false, b1,
                                                          (short)0, acc11, false, false);
        }
    }

    // ---- store D: VGPR r of tile (mi,ni) -> row mi*16 + 8*half + r ----
    const int rbase = tileM + wm * 32 + (half << 3);
    const int cbase = tileN + wn * 32 + l;
    const float bv0 = bias ? bias[cbase]      : 0.0f;
    const float bv1 = bias ? bias[cbase + 16] : 0.0f;
#pragma unroll
    for (int r = 0; r < 8; ++r) {
        const size_t r0 = (size_t)(rbase + r) * ldc;
        const size_t r1 = (size_t)(rbase + 16 + r) * ldc;
        C[r0 + cbase]      = alpha * acc00[r] + bv0;
        C[r0 + cbase + 16] = alpha * acc01[r] + bv1;
        C[r1 + cbase]      = alpha * acc10[r] + bv0;
        C[r1 + cbase + 16] = alpha * acc11[r] + bv1;
    }
}

// ---------------------------------------------------------------------------
// Kernels
// ---------------------------------------------------------------------------

// C[M,N] = X[M,K] @ W[N,K]^T + bias.  N = gridDim.y * BN, M = gridDim.x * BM.
__global__ __launch_bounds__(256)
void proj_gemm_kernel(const float* __restrict__ X, const float* __restrict__ W,
                      const float* __restrict__ bias, float* __restrict__ C, int K)
{
    using namespace cfg;
    const int N = gridDim.y * BN;
    gemm_core<true>(X, K, W, K, bias, C, N, K, 1.0f,
                    blockIdx.x * BM, blockIdx.y * BN);
}

// scores[bh] = Q[b,:,h*DK:+DK] @ Kp[b,:,g*DK:+DK]^T * 1/sqrt(DK)
__global__ __launch_bounds__(256)
void scores_gemm_kernel(const float* __restrict__ Q, const float* __restrict__ Kp,
                        float* __restrict__ attn)
{
    using namespace cfg;
    const int z = blockIdx.z;            // b*H + h
    const int b = z / H;
    const int h = z % H;
    const int g = h / REP;
    const float* Ap = Q  + (size_t)b * S * DM  + h * DK;
    const float* Bp = Kp + (size_t)b * S * KVD + g * DK;
    float*       Cp = attn + (size_t)z * S * S;
    gemm_core<true>(Ap, DM, Bp, KVD, nullptr, Cp, S, DK, 0.125f,
                    blockIdx.x * BM, blockIdx.y * BN);
}

// Row-wise softmax in place over attn rows of length S. One block per row.
__global__ __launch_bounds__(256)
void softmax_rows_kernel(float* __restrict__ attn)
{
    using namespace cfg;
    __shared__ float red[256];
    float* p = attn + (size_t)blockIdx.x * S;
    const int t = threadIdx.x;

    float v[8];
    float m = -3.402823466e+38f;
#pragma unroll
    for (int i = 0; i < 8; ++i) {
        v[i] = p[t + i * 256];
        m = fmaxf(m, v[i]);
    }
    red[t] = m;
    __syncthreads();
#pragma unroll
    for (int s = 128; s > 0; s >>= 1) {
        if (t < s) red[t] = fmaxf(red[t], red[t + s]);
        __syncthreads();
    }
    m = red[0];
    __syncthreads();

    float sum = 0.0f;
#pragma unroll
    for (int i = 0; i < 8; ++i) {
        v[i] = __expf(v[i] - m);
        sum += v[i];
    }
    red[t] = sum;
    __syncthreads();
#pragma unroll
    for (int s = 128; s > 0; s >>= 1) {
        if (t < s) red[t] += red[t + s];
        __syncthreads();
    }
    const float inv = 1.0f / red[0];
#pragma unroll
    for (int i = 0; i < 8; ++i) p[t + i * 256] = v[i] * inv;
}

// ctx[b,:,h*DK:+DK] = attn[bh] @ Vp[b,:,g*DK:+DK]   (NN gemm, K = S)
__global__ __launch_bounds__(256)
void ctx_gemm_kernel(const float* __restrict__ attn, const float* __restrict__ Vp,
                     float* __restrict__ ctx)
{
    using namespace cfg;
    const int z = blockIdx.z;
    const int b = z / H;
    const int h = z % H;
    const int g = h / REP;
    const float* Ap = attn + (size_t)z * S * S;
    const float* Bp = Vp + (size_t)b * S * KVD + g * DK;
    float*       Cp = ctx + (size_t)b * S * DM + h * DK;
    gemm_core<false>(Ap, S, Bp, KVD, nullptr, Cp, DM, S, 1.0f,
                     blockIdx.x * BM, blockIdx.y * BN);
}

// ---------------------------------------------------------------------------
// Host launcher
// ---------------------------------------------------------------------------
extern "C" void kernel_launch(void* const* d_in, const int* in_sizes, int n_in,
                              void* d_out, int out_size, void* d_ws, size_t ws_size,
                              hipStream_t stream)
{
    using namespace cfg;
    const float* query = (const float*)d_in[0];
    const float* key   = (const float*)d_in[1];
    const float* value = (const float*)d_in[2];
    const float* Wq    = (const float*)d_in[3];
    const float* bq    = (const float*)d_in[4];
    const float* Wk    = (const float*)d_in[5];
    const float* bk    = (const float*)d_in[6];
    const float* Wv    = (const float*)d_in[7];
    const float* bv    = (const float*)d_in[8];
    const float* Wo    = (const float*)d_in[9];
    const float* bo    = (const float*)d_in[10];

    float* out  = (float*)d_out;                       // [B,S,DM]
    float* attn = out + (size_t)B * S * DM;            // [B,H,S,S]

    float* ws  = (float*)d_ws;
    float* Q   = ws;                                   // [B,S,DM]   16 MB
    float* Kp  = Q  + (size_t)B * S * DM;              // [B,S,KVD]   4 MB
    float* Vp  = Kp + (size_t)B * S * KVD;             // [B,S,KVD]   4 MB
    float* CTX = Vp + (size_t)B * S * KVD;             // [B,S,DM]   16 MB

    const dim3 blk(256);
    const int M = B * S;                               // 4096

    // 1) projections
    proj_gemm_kernel<<<dim3(M / BM, DM  / BN), blk, 0, stream>>>(query, Wq, bq, Q,  DM);
    proj_gemm_kernel<<<dim3(M / BM, KVD / BN), blk, 0, stream>>>(key,   Wk, bk, Kp, DM);
    proj_gemm_kernel<<<dim3(M / BM, KVD / BN), blk, 0, stream>>>(value, Wv, bv, Vp, DM);

    // 2) scores -> raw logits written into attn region of d_out
    scores_gemm_kernel<<<dim3(S / BM, S / BN, B * H), blk, 0, stream>>>(Q, Kp, attn);

    // 3) softmax in place (one block per attn row)
    softmax_rows_kernel<<<dim3(B * H * S), blk, 0, stream>>>(attn);

    // 4) ctx = attn @ V, stored directly in [b, s, H*DK] layout
    ctx_gemm_kernel<<<dim3(S / BM, DK / BN, B * H), blk, 0, stream>>>(attn, Vp, CTX);

    // 5) output projection
    proj_gemm_kernel<<<dim3(M / BM, DM / BN), blk, 0, stream>>>(CTX, Wo, bo, out, DM);
}